// HeteroGNN_56899726737797
// MI455X (gfx1250) — compile-verified
//
#include <hip/hip_runtime.h>
#include <hip/hip_bf16.h>

typedef float v2f  __attribute__((ext_vector_type(2)));
typedef float v8f  __attribute__((ext_vector_type(8)));

#define SLOPE 0.2f
#define KC 64            // K-chunk staged in LDS (all our K are multiples of 64)

// ---------- helpers ----------
__device__ __forceinline__ unsigned f2ord(float f) {
    unsigned u = __float_as_uint(f);
    return (u & 0x80000000u) ? ~u : (u | 0x80000000u);
}
__device__ __forceinline__ float ord2f(unsigned u) {
    unsigned v = (u & 0x80000000u) ? (u & 0x7fffffffu) : ~u;
    return __uint_as_float(v);
}

// ---------- WMMA GEMM: C[M,N] = A[M,K] @ B[K,N], row-major fp32 ----------
// Compile-time N,K => all B loads use immediate offsets (no per-step addr math).
// Each block stages a 32-row A tile in LDS; each wave owns one 16-col n-tile and
// computes TWO 16x16 m-tiles, so every B fragment feeds two independent WMMAs.
// Requires N%16==0, K%KC==0, gridDim.x == ceil(M/32), gridDim.y*nw == N/16.
template <int N, int K>
__global__ void gemm_wmma(const float* __restrict__ A, const float* __restrict__ B,
                          float* __restrict__ C, int M) {
#if defined(__HIP_DEVICE_COMPILE__) && __has_builtin(__builtin_amdgcn_wmma_f32_16x16x4_f32)
    __shared__ float As[32 * 68];                 // 68-float row pitch: conflict-free frag reads
    const int wib  = threadIdx.x >> 5;
    const int nw   = blockDim.x >> 5;
    const int m0   = blockIdx.x << 5;             // 32 rows per block
    const int n0   = (blockIdx.y * nw + wib) << 4;
    const int lane = threadIdx.x & 31;
    const int half = lane >> 4;                   // 0: lanes 0-15, 1: lanes 16-31
    const int lr   = lane & 15;

    v8f acc0 = {}, acc1 = {};
    for (int k0 = 0; k0 < K; k0 += KC) {
        __syncthreads();
        // stage A[m0:m0+32, k0:k0+KC] -> LDS; clamp OOB rows (discarded at store)
        for (int u = threadIdx.x; u < 512; u += blockDim.x) {
            int r = u >> 4, c = (u & 15) << 2;
            int mr = m0 + r; if (mr >= M) mr = M - 1;
            float4 v = *(const float4*)(A + (size_t)mr * K + k0 + c);
            As[r * 68 + c + 0] = v.x; As[r * 68 + c + 1] = v.y;
            As[r * 68 + c + 2] = v.z; As[r * 68 + c + 3] = v.w;
        }
        __syncthreads();
        const float* Ap = As + lr * 68 + half * 2;            // m-tile 0
        const float* Bp = B + (size_t)(k0 + half * 2) * N + n0 + lr;
        #pragma unroll
        for (int kk = 0; kk < KC; kk += 4) {
            v2f a0 = *(const v2f*)(Ap + kk);                  // rows lr
            v2f a1 = *(const v2f*)(Ap + 16 * 68 + kk);        // rows lr+16
            v2f b;
            b.x = Bp[kk * N];                                 // B[ka][n0+lr]
            b.y = Bp[kk * N + N];                             // B[ka+1][n0+lr]
            acc0 = __builtin_amdgcn_wmma_f32_16x16x4_f32(false, a0, false, b,
                                                         (short)0, acc0, false, false);
            acc1 = __builtin_amdgcn_wmma_f32_16x16x4_f32(false, a1, false, b,
                                                         (short)0, acc1, false, false);
        }
    }
    // C/D layout: VGPR i, lanes 0-15 -> M=m0+i, lanes 16-31 -> M=m0+8+i
    #pragma unroll
    for (int i = 0; i < 8; ++i) {
        int mr0 = m0 + half * 8 + i;
        if (mr0 < M) C[(size_t)mr0 * N + n0 + lr] = acc0[i];
        int mr1 = m0 + 16 + half * 8 + i;
        if (mr1 < M) C[(size_t)mr1 * N + n0 + lr] = acc1[i];
    }
#endif
}

// ---------- fold: V[k,h] = sum_c W[k, h*C+c] * a[h,c] ----------
__global__ void fold_wa(const float* __restrict__ W, const float* __restrict__ Av,
                        float* __restrict__ V, int K, int H, int C) {
    int idx = blockIdx.x * blockDim.x + threadIdx.x;
    if (idx >= K * H) return;
    int k = idx / H, h = idx % H;
    int HCd = H * C;
    float s = 0.f;
    for (int c = 0; c < C; ++c)
        s = fmaf(W[(size_t)k * HCd + h * C + c], Av[h * C + c], s);
    V[idx] = s;
}

// ---------- per-node attention scores, wave-per-node (coalesced) ----------
__global__ void score_gemv(const float* __restrict__ X, const float* __restrict__ V,
                           float* __restrict__ S, int N, int K, int H) {
    int nwv = blockDim.x >> 5;
    int n = blockIdx.x * nwv + (threadIdx.x >> 5);
    if (n >= N) return;
    int lane = threadIdx.x & 31;
    float acc[4] = {0.f, 0.f, 0.f, 0.f};
    for (int k = lane; k < K; k += 32) {
        float xv = X[(size_t)n * K + k];
        for (int h = 0; h < H; ++h) acc[h] = fmaf(xv, V[k * H + h], acc[h]);
    }
    for (int off = 16; off > 0; off >>= 1)
        for (int h = 0; h < H; ++h) acc[h] += __shfl_xor(acc[h], off, 32);
    if (lane == 0)
        for (int h = 0; h < H; ++h) S[(size_t)n * H + h] = acc[h];
}

// ---------- edge pass 1: e = leaky_relu(es[src]+ed[dst]); segment max ----------
__global__ void edge_pass1(const int* __restrict__ src, const int* __restrict__ dst,
                           const float* __restrict__ es, const float* __restrict__ ed,
                           float* __restrict__ ebuf, unsigned* __restrict__ mflip,
                           int E, int H) {
    int i = blockIdx.x * blockDim.x + threadIdx.x;
    if (i >= E) return;
    int s = src[i], d = dst[i];
    for (int h = 0; h < H; ++h) {
        float e = es[(size_t)s * H + h] + ed[(size_t)d * H + h];
        e = e > 0.f ? e : SLOPE * e;
        ebuf[(size_t)i * H + h] = e;
        atomicMax(&mflip[(size_t)d * H + h], f2ord(e));
    }
}

// ---------- edge pass 2: w = exp(e - m[dst]); segment sum ----------
__global__ void edge_pass2(const int* __restrict__ dst, float* __restrict__ ebuf,
                           const unsigned* __restrict__ mflip, float* __restrict__ z,
                           int E, int H) {
    int i = blockIdx.x * blockDim.x + threadIdx.x;
    if (i >= E) return;
    int d = dst[i];
    for (int h = 0; h < H; ++h) {
        float m = ord2f(mflip[(size_t)d * H + h]);
        float w = __expf(ebuf[(size_t)i * H + h] - m);
        ebuf[(size_t)i * H + h] = w;
        atomicAdd(&z[(size_t)d * H + h], w);
    }
}

// ---------- edge pass 3: out[dst, h*C+c] += hs[src, h*C+c] * (w/z[dst,h]) ----------
// blockDim.x == H*C; block iterates edges grid-stride, threads cover channels.
__global__ void edge_pass3(const int* __restrict__ src, const int* __restrict__ dst,
                           const float* __restrict__ wbuf, const float* __restrict__ z,
                           const float* __restrict__ hs, float* __restrict__ out,
                           int E, int H, int C) {
    int tid = threadIdx.x;
    int h   = tid / C;
    int HCd = H * C;
    for (int i = blockIdx.x; i < E; i += gridDim.x) {
        int s = src[i], d = dst[i];
        float alpha = wbuf[(size_t)i * H + h] / z[(size_t)d * H + h];
        atomicAdd(&out[(size_t)d * HCd + tid], hs[(size_t)s * HCd + tid] * alpha);
    }
}

// ---------- bias (+optional second bias) + relu, in place ----------
__global__ void bias_relu(float* __restrict__ buf, const float* __restrict__ b0,
                          const float* __restrict__ b1p, int N, int HCd) {
    size_t i = (size_t)blockIdx.x * blockDim.x + threadIdx.x;
    if (i >= (size_t)N * HCd) return;
    int j = (int)(i % HCd);
    float v = buf[i] + b0[j] + (b1p ? b1p[j] : 0.f);
    buf[i] = v > 0.f ? v : 0.f;
}

// ---------- final: logits = emp2 @ wl + bl; log_softmax; wave-per-node ----------
__global__ void logits_logsoftmax(const float* __restrict__ emp2,
                                  const float* __restrict__ wl,
                                  const float* __restrict__ bl,
                                  float* __restrict__ out, int N) {
    int nwv = blockDim.x >> 5;
    int n = blockIdx.x * nwv + (threadIdx.x >> 5);
    if (n >= N) return;
    int lane = threadIdx.x & 31;
    float l[16];
    #pragma unroll
    for (int o = 0; o < 16; ++o) l[o] = 0.f;
    for (int k = lane; k < 64; k += 32) {         // coalesced across lanes
        float xv = emp2[(size_t)n * 64 + k];
        #pragma unroll
        for (int o = 0; o < 16; ++o) l[o] = fmaf(xv, wl[k * 16 + o], l[o]);
    }
    #pragma unroll
    for (int o = 0; o < 16; ++o)
        for (int off = 16; off > 0; off >>= 1) l[o] += __shfl_xor(l[o], off, 32);
    if (lane == 0) {
        #pragma unroll
        for (int o = 0; o < 16; ++o) l[o] += bl[o];
        float mx = l[0];
        #pragma unroll
        for (int o = 1; o < 16; ++o) mx = fmaxf(mx, l[o]);
        float s = 0.f;
        #pragma unroll
        for (int o = 0; o < 16; ++o) s += __expf(l[o] - mx);
        float lse = mx + __logf(s);
        #pragma unroll
        for (int o = 0; o < 16; ++o) out[(size_t)n * 16 + o] = l[o] - lse;
    }
}

extern "C" void kernel_launch(void* const* d_in, const int* in_sizes, int n_in,
                              void* d_out, int out_size, void* d_ws, size_t ws_size,
                              hipStream_t stream) {
    const float* x_emp = (const float*)d_in[0];
    const float* x_dep = (const float*)d_in[1];
    const float* x_tit = (const float*)d_in[2];
    const int* src_wd = (const int*)d_in[3];
    const int* dst_wd = (const int*)d_in[4];
    const int* src_hr = (const int*)d_in[5];
    const int* dst_hr = (const int*)d_in[6];
    const int* src_rw = (const int*)d_in[7];
    const int* dst_rw = (const int*)d_in[8];
    const int* src_rh = (const int*)d_in[9];
    const int* dst_rh = (const int*)d_in[10];
    const float* w1_s = (const float*)d_in[11];  // [4,64,256]
    const float* w1_d = (const float*)d_in[12];
    const float* a1_s = (const float*)d_in[13];  // [4,4,64]
    const float* a1_d = (const float*)d_in[14];
    const float* b1   = (const float*)d_in[15];  // [4,256]
    const float* w2_s = (const float*)d_in[16];  // [4,256,64]
    const float* w2_d = (const float*)d_in[17];
    const float* a2_s = (const float*)d_in[18];  // [4,1,64]
    const float* a2_d = (const float*)d_in[19];
    const float* b2   = (const float*)d_in[20];  // [4,64]
    const float* wl   = (const float*)d_in[21];  // [64,16]
    const float* bl   = (const float*)d_in[22];  // [16]
    float* outp = (float*)d_out;

    const int FIN = 64, HID = 64, HEADS = 4, HC = HEADS * HID;  // 256
    const int NE = in_sizes[0] / FIN;  // 200000
    const int ND = in_sizes[1] / FIN;  // 500
    const int NT = in_sizes[2] / FIN;  // 200
    const int E  = in_sizes[3];        // 200000

    // ---- workspace layout (floats) ----
    float* ws = (float*)d_ws;
    size_t o = 0;
    float* hs1_0 = ws + o; o += (size_t)NE * HC;
    float* hs1_1 = ws + o; o += (size_t)NE * HC;
    float* hs1_2 = ws + o; o += (size_t)512 * HC;
    float* hs1_3 = ws + o; o += (size_t)256 * HC;
    float* dep_out = ws + o; o += (size_t)512 * HC;
    float* tit_out = ws + o; o += (size_t)256 * HC;
    float* emp_out = ws + o; o += (size_t)NE * HC;
    float* emp2    = ws + o; o += (size_t)NE * HID;
    float* es1_0 = ws + o; o += (size_t)NE * 4;
    float* es1_1 = ws + o; o += (size_t)NE * 4;
    float* ed1_2 = ws + o; o += (size_t)NE * 4;
    float* ed1_3 = ws + o; o += (size_t)NE * 4;
    float* ed1_0 = ws + o; o += 512 * 4;
    float* ed1_1 = ws + o; o += 256 * 4;
    float* es1_2 = ws + o; o += 512 * 4;
    float* es1_3 = ws + o; o += 256 * 4;
    float* hs2_2 = ws + o; o += (size_t)512 * HID;
    float* hs2_3 = ws + o; o += (size_t)256 * HID;
    float* es2_2 = ws + o; o += 512;
    float* es2_3 = ws + o; o += 256;
    float* ed2_2 = ws + o; o += (size_t)NE;
    float* ed2_3 = ws + o; o += (size_t)NE;
    float* folds = ws + o; o += 12 * 1024;        // 12 slots of 1024 floats
    float* ew    = ws + o; o += (size_t)E * 4;    // edge weights scratch
    unsigned* mb = (unsigned*)(ws + o); o += (size_t)NE * 4;  // flipped max
    float* zb    = ws + o; o += (size_t)NE * 4;   // softmax denominators

    // N,K compile-time: immediate-offset B loads inside the WMMA loop.
    auto gemm_n256k64 = [&](const float* A, const float* B, float* Cp, int M) {
        dim3 grid((M + 31) / 32, 2);              // ntiles=16, nw=8
        gemm_wmma<256, 64><<<grid, 256, 0, stream>>>(A, B, Cp, M);
    };
    auto gemm_n64k256 = [&](const float* A, const float* B, float* Cp, int M) {
        dim3 grid((M + 31) / 32, 1);              // ntiles=4, nw=4
        gemm_wmma<64, 256><<<grid, 128, 0, stream>>>(A, B, Cp, M);
    };
    auto relpass = [&](const int* s, const int* d, const float* es, const float* ed,
                       const float* hs, float* ob, int ndst, int H, int C) {
        hipMemsetAsync(mb, 0, (size_t)ndst * H * sizeof(unsigned), stream);
        hipMemsetAsync(zb, 0, (size_t)ndst * H * sizeof(float), stream);
        int eb = (E + 255) / 256;
        edge_pass1<<<eb, 256, 0, stream>>>(s, d, es, ed, ew, mb, E, H);
        edge_pass2<<<eb, 256, 0, stream>>>(d, ew, mb, zb, E, H);
        edge_pass3<<<4096, H * C, 0, stream>>>(s, d, ew, zb, hs, ob, E, H, C);
    };

    // ---- folds: V = W · a  (es/ed become simple GEMVs from raw features) ----
    for (int r = 0; r < 4; ++r) {
        fold_wa<<<1, 256, 0, stream>>>(w1_s + (size_t)r * 64 * 256, a1_s + r * 256,
                                       folds + (2 * r + 0) * 1024, 64, 4, 64);
        fold_wa<<<1, 256, 0, stream>>>(w1_d + (size_t)r * 64 * 256, a1_d + r * 256,
                                       folds + (2 * r + 1) * 1024, 64, 4, 64);
    }
    fold_wa<<<1, 256, 0, stream>>>(w2_s + (size_t)2 * 256 * 64, a2_s + 2 * 64, folds + 8 * 1024, 256, 1, 64);
    fold_wa<<<1, 256, 0, stream>>>(w2_d + (size_t)2 * 256 * 64, a2_d + 2 * 64, folds + 9 * 1024, 256, 1, 64);
    fold_wa<<<1, 256, 0, stream>>>(w2_s + (size_t)3 * 256 * 64, a2_s + 3 * 64, folds + 10 * 1024, 256, 1, 64);
    fold_wa<<<1, 256, 0, stream>>>(w2_d + (size_t)3 * 256 * 64, a2_d + 3 * 64, folds + 11 * 1024, 256, 1, 64);

    // ---- layer-1 source projections (WMMA GEMMs) ----
    gemm_n256k64(x_emp, w1_s + (size_t)0 * 64 * 256, hs1_0, NE);
    gemm_n256k64(x_emp, w1_s + (size_t)1 * 64 * 256, hs1_1, NE);
    gemm_n256k64(x_dep, w1_s + (size_t)2 * 64 * 256, hs1_2, ND);
    gemm_n256k64(x_tit, w1_s + (size_t)3 * 64 * 256, hs1_3, NT);

    // ---- layer-1 attention scores (wave per node, 8 nodes/block) ----
    int gNE = (NE + 7) / 8, gND = (ND + 7) / 8, gNT = (NT + 7) / 8;
    score_gemv<<<gNE, 256, 0, stream>>>(x_emp, folds + 0 * 1024, es1_0, NE, 64, 4);
    score_gemv<<<gND, 256, 0, stream>>>(x_dep, folds + 1 * 1024, ed1_0, ND, 64, 4);
    score_gemv<<<gNE, 256, 0, stream>>>(x_emp, folds + 2 * 1024, es1_1, NE, 64, 4);
    score_gemv<<<gNT, 256, 0, stream>>>(x_tit, folds + 3 * 1024, ed1_1, NT, 64, 4);
    score_gemv<<<gND, 256, 0, stream>>>(x_dep, folds + 4 * 1024, es1_2, ND, 64, 4);
    score_gemv<<<gNE, 256, 0, stream>>>(x_emp, folds + 5 * 1024, ed1_2, NE, 64, 4);
    score_gemv<<<gNT, 256, 0, stream>>>(x_tit, folds + 6 * 1024, es1_3, NT, 64, 4);
    score_gemv<<<gNE, 256, 0, stream>>>(x_emp, folds + 7 * 1024, ed1_3, NE, 64, 4);

    // ---- layer-1 message passing ----
    hipMemsetAsync(dep_out, 0, (size_t)512 * HC * sizeof(float), stream);
    hipMemsetAsync(tit_out, 0, (size_t)256 * HC * sizeof(float), stream);
    hipMemsetAsync(emp_out, 0, (size_t)NE * HC * sizeof(float), stream);
    relpass(src_wd, dst_wd, es1_0, ed1_0, hs1_0, dep_out, ND, 4, 64);
    relpass(src_hr, dst_hr, es1_1, ed1_1, hs1_1, tit_out, NT, 4, 64);
    relpass(src_rw, dst_rw, es1_2, ed1_2, hs1_2, emp_out, NE, 4, 64);
    relpass(src_rh, dst_rh, es1_3, ed1_3, hs1_3, emp_out, NE, 4, 64);

    bias_relu<<<((size_t)ND * HC + 255) / 256, 256, 0, stream>>>(dep_out, b1 + 0 * HC, nullptr, ND, HC);
    bias_relu<<<((size_t)NT * HC + 255) / 256, 256, 0, stream>>>(tit_out, b1 + 1 * HC, nullptr, NT, HC);
    bias_relu<<<((size_t)NE * HC + 255) / 256, 256, 0, stream>>>(emp_out, b1 + 2 * HC, b1 + 3 * HC, NE, HC);

    // ---- layer 2 (heads=1) ----
    gemm_n64k256(dep_out, w2_s + (size_t)2 * 256 * 64, hs2_2, ND);
    gemm_n64k256(tit_out, w2_s + (size_t)3 * 256 * 64, hs2_3, NT);
    score_gemv<<<gND, 256, 0, stream>>>(dep_out, folds + 8 * 1024,  es2_2, ND, 256, 1);
    score_gemv<<<gNE, 256, 0, stream>>>(emp_out, folds + 9 * 1024,  ed2_2, NE, 256, 1);
    score_gemv<<<gNT, 256, 0, stream>>>(tit_out, folds + 10 * 1024, es2_3, NT, 256, 1);
    score_gemv<<<gNE, 256, 0, stream>>>(emp_out, folds + 11 * 1024, ed2_3, NE, 256, 1);

    hipMemsetAsync(emp2, 0, (size_t)NE * HID * sizeof(float), stream);
    relpass(src_rw, dst_rw, es2_2, ed2_2, hs2_2, emp2, NE, 1, 64);
    relpass(src_rh, dst_rh, es2_3, ed2_3, hs2_3, emp2, NE, 1, 64);
    bias_relu<<<((size_t)NE * HID + 255) / 256, 256, 0, stream>>>(emp2, b2 + 2 * HID, b2 + 3 * HID, NE, HID);

    // ---- classifier + log_softmax (wave per node) ----
    logits_logsoftmax<<<gNE, 256, 0, stream>>>(emp2, wl, bl, outp, NE);
}